// RobustLoss_2345052143825
// MI455X (gfx1250) — compile-verified
//
#include <hip/hip_runtime.h>
#include <math.h>

// ---------------- problem constants ----------------
#define BLOCK   256
#define NB      1024          // grid blocks for streaming kernels
#define EPT     64            // elements per thread: NB*BLOCK*EPT == 2^24
#define NCH     (EPT / 4)     // float4 chunks per thread
#define ETAS    16            // etas evaluated per pass (multisection)
#define ROUNDS  8             // multisection rounds: bracket /17^8

#define REGC     0.1f
#define INV_REG  10.0f
#define INV_SIZE 10.0f        // 1/SIZE
#define TOLC     1e-4f
#define INV_M    (1.0f/16777216.0f)
#define LOG_M    16.635532333438686f
#define SHIFT    40.0f        // overflow guard for logsumexp pass

// ---------------- workspace layout (floats) ----------------
#define P16_OFF  0                   // NB*ETAS per-block partial sums
#define P2_OFF   (NB*ETAS)           // NB*2   per-block (max, sumexp)
#define CTRL_OFF (NB*ETAS + NB*2)    // control block
// ctrl: [0]=lo [1]=hi [2]=done [3]=eta_star [4]=eta_min [5]=eta_max

typedef float v2f __attribute__((ext_vector_type(2)));
typedef float v8f __attribute__((ext_vector_type(8)));
typedef int   v4i __attribute__((ext_vector_type(4)));

// --------- optional CDNA5 async Global->LDS path (guarded; cannot break build) ---------
#if __has_builtin(__builtin_amdgcn_global_load_async_to_lds_b128) && \
    __has_builtin(__builtin_amdgcn_s_wait_asynccnt)
#define HAVE_ASYNC_LDS 1
typedef __attribute__((address_space(1))) v4i g_v4i;   // global (device) int4
typedef __attribute__((address_space(3))) v4i l_v4i;   // LDS int4
#else
#define HAVE_ASYNC_LDS 0
#endif

__device__ __forceinline__ float wave_sum(float x) {
  #pragma unroll
  for (int off = 16; off; off >>= 1) x += __shfl_down(x, off, 32);
  return x;
}
__device__ __forceinline__ float wave_max(float x) {
  #pragma unroll
  for (int off = 16; off; off >>= 1) x = fmaxf(x, __shfl_down(x, off, 32));
  return x;
}

// ---- pass 1: per-block max(v) and sum exp(v/REG - SHIFT) (HBM-bound pass) ----
__global__ __launch_bounds__(BLOCK) void k_stats(const float* __restrict__ v,
                                                 float* __restrict__ W) {
  const int tid = threadIdx.x, b = blockIdx.x;
  const float4* vv = (const float4*)(v + (size_t)b * (BLOCK * EPT));
  float m = -3.4e38f, s = 0.f;
  for (int k = 0; k < NCH; ++k) {
    // CDNA5 speculative prefetch (global_prefetch_b8) a few tiles ahead
    __builtin_prefetch((const char*)(vv + (k + 4) * BLOCK + tid), 0, 1);
    float4 x = vv[k * BLOCK + tid];
    m = fmaxf(m, fmaxf(fmaxf(x.x, x.y), fmaxf(x.z, x.w)));
    s += __expf(x.x * INV_REG - SHIFT) + __expf(x.y * INV_REG - SHIFT)
       + __expf(x.z * INV_REG - SHIFT) + __expf(x.w * INV_REG - SHIFT);
  }
  __shared__ float shm[BLOCK / 32], shs[BLOCK / 32];
  const int lane = tid & 31, wid = tid >> 5;
  float wm = wave_max(m), ws_ = wave_sum(s);
  if (!lane) { shm[wid] = wm; shs[wid] = ws_; }
  __syncthreads();
  if (!tid) {
    float M2 = shm[0], S2 = shs[0];
    for (int w = 1; w < BLOCK / 32; ++w) { M2 = fmaxf(M2, shm[w]); S2 += shs[w]; }
    W[P2_OFF + b * 2 + 0] = M2;
    W[P2_OFF + b * 2 + 1] = S2;
  }
}

// ---- combine stats -> eta_min (logsumexp), eta_max (max), init control ----
__global__ __launch_bounds__(BLOCK) void k_stats_combine(float* __restrict__ W) {
  const int tid = threadIdx.x;
  float m = -3.4e38f, s = 0.f;
  for (int i = tid; i < NB; i += BLOCK) {
    m = fmaxf(m, W[P2_OFF + i * 2 + 0]);
    s += W[P2_OFF + i * 2 + 1];
  }
  __shared__ float shm[BLOCK / 32], shs[BLOCK / 32];
  const int lane = tid & 31, wid = tid >> 5;
  float wm = wave_max(m), ws_ = wave_sum(s);
  if (!lane) { shm[wid] = wm; shs[wid] = ws_; }
  __syncthreads();
  if (!tid) {
    float M2 = shm[0], S2 = shs[0];
    for (int w = 1; w < BLOCK / 32; ++w) { M2 = fmaxf(M2, shm[w]); S2 += shs[w]; }
    float eta_min = REGC * (logf(S2) + SHIFT - LOG_M);
    float* C = W + CTRL_OFF;
    C[0] = eta_min; C[1] = M2; C[2] = 0.f;
    C[3] = 0.5f * (eta_min + M2); C[4] = eta_min; C[5] = M2;
  }
}

// per-element update of 16 running sums: min(exp(v/REG)*t_j, 1/SIZE)
__device__ __forceinline__ void accum16(const float4 x, const float* __restrict__ t,
                                        float* __restrict__ acc) {
  float e0 = fmaxf(__expf(x.x * INV_REG), 1e-37f);
  float e1 = fmaxf(__expf(x.y * INV_REG), 1e-37f);
  float e2 = fmaxf(__expf(x.z * INV_REG), 1e-37f);
  float e3 = fmaxf(__expf(x.w * INV_REG), 1e-37f);
  #pragma unroll
  for (int j = 0; j < ETAS; ++j)
    acc[j] += fminf(e0 * t[j], INV_SIZE) + fminf(e1 * t[j], INV_SIZE)
            + fminf(e2 * t[j], INV_SIZE) + fminf(e3 * t[j], INV_SIZE);
}

// ---- streaming evaluation of f at 16 etas (mode 0: expansion pts, 1: multisection) ----
__global__ __launch_bounds__(BLOCK) void k_eval(const float* __restrict__ v,
                                                float* __restrict__ W, int mode) {
  const int tid = threadIdx.x, b = blockIdx.x;
  const float* C = W + CTRL_OFF;
  float t[ETAS];
  if (mode == 0) {  // geometric bracket-expansion candidates (mirrors reference loop)
    const float emin = C[4], emax = C[5], L = emax - emin;
    #pragma unroll
    for (int j = 0; j < ETAS; ++j) {
      float eta = (j == 0) ? emin : emin - (float)((1u << (j + 1)) - 2u) * L;
      t[j] = __expf(-eta * INV_REG);   // inf is fine: min(E*inf,10)=10
    }
  } else {          // 16-way multisection of current bracket
    const float lo = C[0], hi = C[1], step = (hi - lo) * (1.0f / 17.0f);
    #pragma unroll
    for (int j = 0; j < ETAS; ++j) t[j] = __expf(-(lo + step * (float)(j + 1)) * INV_REG);
  }
  float acc[ETAS];
  #pragma unroll
  for (int j = 0; j < ETAS; ++j) acc[j] = 0.f;

  const float4* vv = (const float4*)(v + (size_t)b * (BLOCK * EPT));

#if HAVE_ASYNC_LDS
  // Double-buffered async Global->LDS staging (ASYNCcnt pipeline). Each thread
  // stages and consumes only its OWN 16B slot, so no block barriers are needed;
  // async loads complete in order, so s_wait_asynccnt 1 == previous chunk landed.
  __shared__ float4 lbuf[2][BLOCK];
  __builtin_amdgcn_global_load_async_to_lds_b128(
      (g_v4i*)(vv + tid), (l_v4i*)(&lbuf[0][tid]), 0, 0);
  for (int k = 0; k < NCH; ++k) {
    if (k + 1 < NCH) {
      __builtin_amdgcn_global_load_async_to_lds_b128(
          (g_v4i*)(vv + (k + 1) * BLOCK + tid),
          (l_v4i*)(&lbuf[(k + 1) & 1][tid]), 0, 0);
      __builtin_amdgcn_s_wait_asynccnt(1);
    } else {
      __builtin_amdgcn_s_wait_asynccnt(0);
    }
    asm volatile("" ::: "memory");   // compiler fence: LDS written by async op
    float4 x = lbuf[k & 1][tid];
    accum16(x, t, acc);
  }
#else
  for (int k = 0; k < NCH; ++k) {
    float4 x = vv[k * BLOCK + tid];
    accum16(x, t, acc);
  }
#endif

  __shared__ float sh[(BLOCK / 32) * ETAS];
  const int lane = tid & 31, wid = tid >> 5;
  #pragma unroll
  for (int j = 0; j < ETAS; ++j) {
    float s = wave_sum(acc[j]);
    if (!lane) sh[wid * ETAS + j] = s;
  }
  __syncthreads();
  if (tid < ETAS) {
    float s = 0.f;
    for (int w = 0; w < BLOCK / 32; ++w) s += sh[w * ETAS + tid];
    W[P16_OFF + b * ETAS + tid] = s;
  }
}

// ---- WMMA cross-block reduction + bracket / output logic (1 wave) ----
// ones(16x4) x B(4x16) + C accumulates COLUMN sums of B; all-ones A makes the
// result invariant to the B row striping, so this is layout-robust.
__global__ __launch_bounds__(32) void k_combine(float* __restrict__ W, int mode,
                                                float* __restrict__ out) {
  const int lane = threadIdx.x;
  __shared__ float S[ETAS];
#if __has_builtin(__builtin_amdgcn_wmma_f32_16x16x4_f32)
  {
    v2f a; a.x = 1.f; a.y = 1.f;          // A = ones 16x4
    v8f c = {};                            // accumulator 16x16
    const float* P = W + P16_OFF;
    for (int b = 0; b < NB; b += 4) {      // uniform loop, EXEC all-ones
      const int r0 = b + (lane >> 4);
      v2f bm;
      bm.x = P[r0 * ETAS + (lane & 15)];
      bm.y = P[(r0 + 2) * ETAS + (lane & 15)];
      c = __builtin_amdgcn_wmma_f32_16x16x4_f32(false, a, false, bm,
                                                (short)0, c, false, false);
    }
    S[lane & 15] = c[0];                   // D row 0: lane n holds column-sum n
  }
#else
  if (lane < ETAS) {
    float s = 0.f; const float* P = W + P16_OFF;
    for (int b = 0; b < NB; ++b) s += P[b * ETAS + lane];
    S[lane] = s;
  }
#endif
  __syncthreads();
  if (lane == 0) {
    float* C = W + CTRL_OFF;
    if (mode == 2) {
      // S[0]=dot(p,v), S[1]=sum p log p
      out[0] = S[0] - REGC * (LOG_M + S[1]);
    } else if (mode == 0) {
      const float emin = C[4], emax = C[5], L = emax - emin;
      const float f0 = 1.f - S[0] * INV_M;     // f(eta_min)
      if (fabsf(f0) <= TOLC) { C[2] = 1.f; C[3] = emin; }          // early return p(eta_min)
      else if (f0 < 0.f)     { C[0] = emin; C[1] = emax; C[3] = 0.5f * (emin + emax); }
      else {
        int k = ETAS - 1;
        for (int j = 1; j < ETAS; ++j)
          if (1.f - S[j] * INV_M < 0.f) { k = j; break; }
        float lo = emin - (float)((1u << (k + 1)) - 2u) * L;
        float hi = (k == 1) ? emin : emin - (float)((1u << k) - 2u) * L;
        C[0] = lo; C[1] = hi; C[3] = 0.5f * (lo + hi);
      }
    } else {  // mode 1: multisection update
      if (C[2] == 0.f) {
        const float lo = C[0], hi = C[1], step = (hi - lo) * (1.0f / 17.0f);
        int best = -1, klast = -1; float bv = 1e30f;
        for (int j = 0; j < ETAS; ++j) {
          float f = 1.f - S[j] * INV_M;
          float af = fabsf(f);
          if (af <= TOLC && af < bv) { best = j; bv = af; }
          if (f < 0.f) klast = j;    // f increasing in j: last negative = boundary
        }
        if (best >= 0) { C[2] = 1.f; C[3] = lo + step * (float)(best + 1); }
        else {
          float nlo = (klast < 0) ? lo : lo + step * (float)(klast + 1);
          float nhi = (klast == ETAS - 1) ? hi : lo + step * (float)(klast + 2);
          C[0] = nlo; C[1] = nhi; C[3] = 0.5f * (nlo + nhi);
        }
      }
    }
  }
}

// ---- final pass: dot(p,v) and sum p log p at eta_star ----
__global__ __launch_bounds__(BLOCK) void k_final(const float* __restrict__ v,
                                                 float* __restrict__ W) {
  const int tid = threadIdx.x, b = blockIdx.x;
  const float* C = W + CTRL_OFF;
  const float t = __expf(-C[3] * INV_REG);
  float accd = 0.f, accl = 0.f;
  const float4* vv = (const float4*)(v + (size_t)b * (BLOCK * EPT));
  for (int k = 0; k < NCH; ++k) {
    float4 x = vv[k * BLOCK + tid];
    float xs[4] = {x.x, x.y, x.z, x.w};
    #pragma unroll
    for (int i = 0; i < 4; ++i) {
      float e = fmaxf(__expf(xs[i] * INV_REG), 1e-37f);
      float p = fminf(e * t, INV_SIZE) * INV_M;
      accd += p * xs[i];
      accl += (p > 0.f) ? p * __logf(p) : 0.f;   // mirror p>0 masking
    }
  }
  __shared__ float sh[(BLOCK / 32) * 2];
  const int lane = tid & 31, wid = tid >> 5;
  float sd = wave_sum(accd), sl = wave_sum(accl);
  if (!lane) { sh[wid * 2 + 0] = sd; sh[wid * 2 + 1] = sl; }
  __syncthreads();
  if (tid < 2) {
    float s = 0.f;
    for (int w = 0; w < BLOCK / 32; ++w) s += sh[w * 2 + tid];
    W[P16_OFF + b * ETAS + tid] = s;   // columns 0,1 of the partial matrix
  }
}

extern "C" void kernel_launch(void* const* d_in, const int* in_sizes, int n_in,
                              void* d_out, int out_size, void* d_ws, size_t ws_size,
                              hipStream_t stream) {
  (void)in_sizes; (void)n_in; (void)out_size; (void)ws_size;
  const float* v = (const float*)d_in[0];
  float* W = (float*)d_ws;
  float* out = (float*)d_out;

  k_stats<<<NB, BLOCK, 0, stream>>>(v, W);            // HBM pass: max + logsumexp partials
  k_stats_combine<<<1, BLOCK, 0, stream>>>(W);        // eta_min / eta_max
  k_eval<<<NB, BLOCK, 0, stream>>>(v, W, 0);          // bracket expansion: 16 pts, 1 pass
  k_combine<<<1, 32, 0, stream>>>(W, 0, out);
  for (int r = 0; r < ROUNDS; ++r) {                  // 16-way multisection: /17 per pass
    k_eval<<<NB, BLOCK, 0, stream>>>(v, W, 1);
    k_combine<<<1, 32, 0, stream>>>(W, 1, out);
  }
  k_final<<<NB, BLOCK, 0, stream>>>(v, W);            // dot(p,v), sum p log p
  k_combine<<<1, 32, 0, stream>>>(W, 2, out);         // scalar output
}